// HungarianMatcher_72447508349681
// MI455X (gfx1250) — compile-verified
//
#include <hip/hip_runtime.h>
#include <hip/hip_bf16.h>
#include <cstddef>

// Problem sizes (fixed by the reference)
#define BN 64
#define QN 900
#define TN 100
#define CN 92

typedef __attribute__((ext_vector_type(2))) float v2f;
typedef __attribute__((ext_vector_type(8))) float v8f;

// ---------------------------------------------------------------------------
// Kernel 1: fused cost matrix.
//   cost[b,q,t] = L1(pred_bbox[b,q], tgt_bbox[b,t])
//               - softmax(logits[b,q])[tgt_labels[b,t]]
//               - GIoU(pred_bbox[b,q], tgt_bbox[b,t])
// One wave32 per 16x16 (q,t) tile. The class-cost gather is expressed as
// softmax(logits) @ onehot(labels) and computed with V_WMMA_F32_16X16X4_F32
// (bit-exact: one-hot columns mean each output is a single product + exact
// zero additions). K = 92 = 23 * 4 steps.
// ---------------------------------------------------------------------------
__global__ __launch_bounds__(32)
void detr_cost_kernel(const float* __restrict__ logits,
                      const float* __restrict__ pbox,
                      const int*   __restrict__ tlab,
                      const float* __restrict__ tbox,
                      float* __restrict__ cost)
{
    __shared__ float sh_prob[16][CN + 1];   // softmax probs, padded stride
    __shared__ float sh_qb[16][4];          // pred boxes (cx,cy,w,h)
    __shared__ float sh_tb[16][4];          // tgt boxes
    __shared__ int   sh_lab[16];            // tgt labels (-1 if OOB)

    const int lane = threadIdx.x;           // 0..31 (wave32)
    const int qt = blockIdx.x, tt = blockIdx.y, b = blockIdx.z;
    const int q0 = qt * 16, t0 = tt * 16;

    // Stage logits rows for this q-tile.
    for (int idx = lane; idx < 16 * CN; idx += 32) {
        int r = idx / CN, c = idx % CN;
        int q = q0 + r;
        sh_prob[r][c] = (q < QN) ? logits[((size_t)b * QN + q) * CN + c] : 0.0f;
    }
    if (lane < 16) {
        int q = q0 + lane;
        int t = t0 + lane;
        for (int d = 0; d < 4; ++d) {
            sh_qb[lane][d] = (q < QN) ? pbox[((size_t)b * QN + q) * 4 + d] : 0.25f;
            sh_tb[lane][d] = (t < TN) ? tbox[((size_t)b * TN + t) * 4 + d] : 0.25f;
        }
        sh_lab[lane] = (t < TN) ? tlab[(size_t)b * TN + t] : -1;
    }
    __syncthreads();

    // Softmax in LDS, one row per lane 0..15.
    if (lane < 16) {
        float mx = -3.4e38f;
        for (int c = 0; c < CN; ++c) mx = fmaxf(mx, sh_prob[lane][c]);
        float s = 0.0f;
        for (int c = 0; c < CN; ++c) {
            float e = expf(sh_prob[lane][c] - mx);
            sh_prob[lane][c] = e;
            s += e;
        }
        float inv = 1.0f / s;
        for (int c = 0; c < CN; ++c) sh_prob[lane][c] *= inv;
    }
    __syncthreads();

    // WMMA F32 16x16x4: A 16x4 (conf), B 4x16 (one-hot labels), acc 16x16.
    // A layout: lane l holds row m=l&15; VGPR comp c holds K = 2*(l>>4) + c.
    const int half = lane >> 4;
    const int mrow = lane & 15;
    const int ncol = lane & 15;
    const int labn = sh_lab[ncol];

    v8f acc = {};
    #pragma unroll 1
    for (int k0 = 0; k0 < CN; k0 += 4) {
        int ka = k0 + 2 * half;
        v2f a, bb;
        a.x  = sh_prob[mrow][ka];
        a.y  = sh_prob[mrow][ka + 1];
        bb.x = (labn == ka)     ? 1.0f : 0.0f;
        bb.y = (labn == ka + 1) ? 1.0f : 0.0f;
        acc = __builtin_amdgcn_wmma_f32_16x16x4_f32(
            /*neg_a=*/false, a, /*neg_b=*/false, bb,
            /*c_mod=*/(short)0, acc, /*reuse_a=*/false, /*reuse_b=*/false);
    }

    // Epilogue: target box/corners fixed per lane (column), 8 rows per lane.
    const float tcx = sh_tb[ncol][0], tcy = sh_tb[ncol][1];
    const float tw  = sh_tb[ncol][2], th  = sh_tb[ncol][3];
    const float t_x0 = tcx - 0.5f * tw, t_y0 = tcy - 0.5f * th;
    const float t_x1 = tcx + 0.5f * tw, t_y1 = tcy + 0.5f * th;
    const float t_area = (t_x1 - t_x0) * (t_y1 - t_y0);
    const int tg = t0 + ncol;

    #pragma unroll
    for (int e = 0; e < 8; ++e) {
        // C/D layout: VGPR e -> M = e (lanes 0-15) / 8+e (lanes 16-31).
        int mr = e + 8 * half;
        int qg = q0 + mr;
        if (qg < QN && tg < TN) {
            float px = sh_qb[mr][0], py = sh_qb[mr][1];
            float pw = sh_qb[mr][2], ph = sh_qb[mr][3];
            float l1 = fabsf(px - tcx) + fabsf(py - tcy) +
                       fabsf(pw - tw)  + fabsf(ph - th);
            float p_x0 = px - 0.5f * pw, p_y0 = py - 0.5f * ph;
            float p_x1 = px + 0.5f * pw, p_y1 = py + 0.5f * ph;
            float p_area = (p_x1 - p_x0) * (p_y1 - p_y0);
            float iw = fmaxf(fminf(p_x1, t_x1) - fmaxf(p_x0, t_x0), 0.0f);
            float ih = fmaxf(fminf(p_y1, t_y1) - fmaxf(p_y0, t_y0), 0.0f);
            float inter = iw * ih;
            float uni = p_area + t_area - inter;
            float iou = inter / uni;
            float ew = fmaxf(fmaxf(p_x1, t_x1) - fminf(p_x0, t_x0), 0.0f);
            float eh = fmaxf(fmaxf(p_y1, t_y1) - fminf(p_y0, t_y0), 0.0f);
            float earea = ew * eh;
            float giou = iou - (earea - uni) / earea;
            cost[((size_t)b * QN + qg) * TN + tg] = l1 - acc[e] - giou;
        }
    }
}

// ---------------------------------------------------------------------------
// Kernel 2: Jonker-Volgenant shortest augmenting path on cost^T (100 x 900),
// one 256-thread block per batch. State lives in LDS; the 23 MB cost matrix
// stays resident in the 192 MB L2 across the repeated column scans.
// ---------------------------------------------------------------------------
__global__ __launch_bounds__(256)
void jv_lsa_kernel(const float* __restrict__ cost,
                   float* __restrict__ row_out,   // sorted query indices (as f32)
                   float* __restrict__ col_out)   // matched target indices (as f32)
{
    const int n = TN, m = QN;
    __shared__ float v[QN + 1];
    __shared__ float minv[QN + 1];
    __shared__ int   p[QN + 1];
    __shared__ int   way[QN + 1];
    __shared__ unsigned char used[QN + 1];
    __shared__ float u[TN + 1];
    __shared__ float rmin[256];
    __shared__ int   ridx[256];
    __shared__ int   sh_j0;
    __shared__ int   qcol[TN];

    const int tid = threadIdx.x;
    const int b = blockIdx.x;
    const float* Cb = cost + (size_t)b * QN * TN;   // Cb[q*TN + t]; costT[i][j] = Cb[j*TN + i]
    const float INF = 1e9f;

    for (int j = tid; j <= m; j += 256) { v[j] = 0.0f; p[j] = 0; }
    for (int i = tid; i <= n; i += 256) u[i] = 0.0f;
    __syncthreads();

    for (int i = 0; i < n; ++i) {
        for (int j = tid; j <= m; j += 256) { minv[j] = INF; used[j] = 0; way[j] = 0; }
        if (tid == 0) { p[0] = i + 1; sh_j0 = 0; }
        __syncthreads();

        while (true) {
            int j0 = sh_j0;                 // uniform across block
            if (p[j0] == 0) break;          // uniform condition
            if (tid == 0) used[j0] = 1;
            __syncthreads();

            int   i0  = p[j0];
            float ui0 = u[i0];

            float best = 3.0e38f;
            int   bestj = 0x7fffffff;
            for (int j = tid + 1; j <= m; j += 256) {
                float mv;
                if (!used[j]) {
                    float cur = Cb[(size_t)(j - 1) * TN + (i0 - 1)] - ui0 - v[j];
                    if (cur < minv[j]) { minv[j] = cur; way[j] = j0; }
                    mv = minv[j];
                } else {
                    mv = INF;               // argmin over masked values, like the reference
                }
                if (mv < best || (mv == best && j < bestj)) { best = mv; bestj = j; }
            }
            rmin[tid] = best; ridx[tid] = bestj;
            __syncthreads();
            for (int s = 128; s > 0; s >>= 1) {
                if (tid < s) {
                    float o = rmin[tid + s]; int oj = ridx[tid + s];
                    if (o < rmin[tid] || (o == rmin[tid] && oj < ridx[tid])) {
                        rmin[tid] = o; ridx[tid] = oj;
                    }
                }
                __syncthreads();
            }
            float delta = rmin[0];
            int   j1    = ridx[0];

            // Dual updates: used columns map to distinct rows -> no conflicts.
            for (int j = tid; j <= m; j += 256) {
                if (used[j]) { u[p[j]] += delta; v[j] -= delta; }
                else if (j >= 1) minv[j] -= delta;
            }
            if (tid == 0) sh_j0 = j1;
            __syncthreads();
        }

        // Augment along the alternating path (serial, tiny).
        if (tid == 0) {
            int jj = sh_j0;
            while (jj != 0) { int jp = way[jj]; p[jj] = p[jp]; jj = jp; }
        }
        __syncthreads();
    }

    // q[i] = column assigned to row i (unique writer per row).
    for (int j = tid + 1; j <= m; j += 256) {
        int r = p[j];
        if (r >= 1) qcol[r - 1] = j - 1;
    }
    __syncthreads();

    // order = argsort(q): rank by counting (values are distinct).
    if (tid < n) {
        int myq = qcol[tid];
        int rank = 0;
        for (int k = 0; k < n; ++k) rank += (qcol[k] < myq) ? 1 : 0;
        row_out[(size_t)b * TN + rank] = (float)myq;   // sorted query index
        col_out[(size_t)b * TN + rank] = (float)tid;   // matched target index
    }
}

// ---------------------------------------------------------------------------
// Launch: cost tiles then per-batch JV solve. Outputs concatenated:
//   [ cost (B*Q*T) | row_ind (B*T) | col_ind (B*T) ]  all as f32.
// ---------------------------------------------------------------------------
extern "C" void kernel_launch(void* const* d_in, const int* in_sizes, int n_in,
                              void* d_out, int out_size, void* d_ws, size_t ws_size,
                              hipStream_t stream) {
    (void)in_sizes; (void)n_in; (void)out_size; (void)d_ws; (void)ws_size;
    const float* logits = (const float*)d_in[0];
    const float* pbox   = (const float*)d_in[1];
    const int*   tlab   = (const int*)d_in[2];
    const float* tbox   = (const float*)d_in[3];

    float* out     = (float*)d_out;
    float* cost    = out;
    float* row_out = out + (size_t)BN * QN * TN;
    float* col_out = row_out + (size_t)BN * TN;

    dim3 g1((QN + 15) / 16, (TN + 15) / 16, BN);   // 57 x 7 x 64 tiles
    detr_cost_kernel<<<g1, 32, 0, stream>>>(logits, pbox, tlab, tbox, cost);
    jv_lsa_kernel<<<BN, 256, 0, stream>>>(cost, row_out, col_out);
}